// mLSTM_54022098649616
// MI455X (gfx1250) — compile-verified
//
#include <hip/hip_runtime.h>
#include <hip/hip_bf16.h>

// ---------------------------------------------------------------------------
// mLSTM step for MI455X (gfx1250, wave32).
//  * All big GEMMs: v_wmma_f32_16x16x4_f32, A/W tiles staged to LDS via
//    GLOBAL_LOAD_ASYNC_TO_LDS_B128 (ASYNCcnt, double-buffered) when the
//    toolchain exposes the builtin; padded LDS rows give bank-conflict-free
//    ds_load_b64 fragment reads.
//  * c-state update (268MB traffic, HBM-bound) streams coalesced float4.
// ---------------------------------------------------------------------------

typedef float v2f __attribute__((ext_vector_type(2)));
typedef float v8f __attribute__((ext_vector_type(8)));
typedef int   v4i __attribute__((ext_vector_type(4)));

#define B_SZ   256
#define DIM_SZ 1024
#define H_SZ   8
#define D_SZ   128
#define P_SZ   2048
#define HD_SZ  1024

#define LDS_STRIDE 68   // 64 + 4 pad dwords -> all-64-bank ds_load_b64 reads

#if defined(__has_builtin)
#if __has_builtin(__builtin_amdgcn_global_load_async_to_lds_b128)
#define USE_ASYNC_LDS 1
#endif
#endif

#ifdef USE_ASYNC_LDS
typedef __attribute__((address_space(1))) v4i GV4;   // global int4
typedef __attribute__((address_space(3))) v4i LV4;   // LDS int4
#endif

__device__ __forceinline__ void async_cp16(const float* gp, float* lp)
{
#ifdef USE_ASYNC_LDS
    __builtin_amdgcn_global_load_async_to_lds_b128(
        (GV4*)(void*)gp, (LV4*)(void*)lp, 0, 0);
#else
    *(float4*)lp = *(const float4*)gp;
#endif
}

template <int N>
__device__ __forceinline__ void wait_async()
{
#ifdef USE_ASYNC_LDS
#if __has_builtin(__builtin_amdgcn_s_wait_asynccnt)
    __builtin_amdgcn_s_wait_asynccnt(N);
#else
    asm volatile("s_wait_asynccnt %0" ::"i"(N) : "memory");
#endif
#endif
}

// ---------------------------------------------------------------------------
// LayerNorm over DIM=1024, one block (8 waves) per row, float4 per thread.
// ---------------------------------------------------------------------------
__global__ void __launch_bounds__(256) layernorm_k(
    const float* __restrict__ x, const float* __restrict__ g,
    const float* __restrict__ bta, float* __restrict__ y)
{
    const int row = blockIdx.x;
    const int tid = threadIdx.x;
    const float4 v = ((const float4*)(x + (size_t)row * DIM_SZ))[tid];
    float s  = v.x + v.y + v.z + v.w;
    float ss = v.x * v.x + v.y * v.y + v.z * v.z + v.w * v.w;
    #pragma unroll
    for (int off = 16; off; off >>= 1) {
        s  += __shfl_xor(s,  off, 32);
        ss += __shfl_xor(ss, off, 32);
    }
    __shared__ float rs[8], rss[8], mb[2];
    const int wave = tid >> 5, lane = tid & 31;
    if (lane == 0) { rs[wave] = s; rss[wave] = ss; }
    __syncthreads();
    if (tid == 0) {
        float S = 0.f, SS = 0.f;
        #pragma unroll
        for (int i = 0; i < 8; ++i) { S += rs[i]; SS += rss[i]; }
        const float mu  = S * (1.f / DIM_SZ);
        const float var = SS * (1.f / DIM_SZ) - mu * mu;
        mb[0] = mu; mb[1] = rsqrtf(var + 1e-5f);
    }
    __syncthreads();
    const float mu = mb[0], inv = mb[1];
    const float4 gg = ((const float4*)g)[tid];
    const float4 bb = ((const float4*)bta)[tid];
    float4 o;
    o.x = (v.x - mu) * inv * gg.x + bb.x;
    o.y = (v.y - mu) * inv * gg.y + bb.y;
    o.z = (v.z - mu) * inv * gg.z + bb.z;
    o.w = (v.w - mu) * inv * gg.w + bb.w;
    ((float4*)(y + (size_t)row * DIM_SZ))[tid] = o;
}

// ---------------------------------------------------------------------------
// WMMA fp32 GEMM:  C[M,N] = (A[M,K] . W[N,K]^T + bias[N]) * scale + resid[M,N]
// Block = 4 waves = 16(M) x 64(N) C tile. K chunked by 64:
//   stage A 16x64 + W 64x64 into LDS (coalesced b128 async copies,
//   double-buffered on ASYNCcnt), then 16 v_wmma_f32_16x16x4_f32 per chunk
//   with conflict-free ds_load_b64 fragment reads.
//
// ISA fragment layouts (CDNA5 ISA):
//   A (16x4): lanes 0-15: M=lane, K={0,1}; lanes 16-31: K={2,3}
//   B (4x16): lanes 0-15: N=lane, K={0,1}; lanes 16-31: K={2,3}
//   C (8 VGPRs): VGPR i -> M = i + (lane>>4)*8, N = lane&15
// ---------------------------------------------------------------------------
__device__ __forceinline__ void stage_chunk(
    const float* __restrict__ A, const float* __restrict__ W,
    int K, int mTile, int nTile0, int kc, float* sA, float* sW, int tid)
{
    // W chunk: 64 rows x 64 floats = 1024 float4, 8 per thread (coalesced)
    #pragma unroll
    for (int i = 0; i < 8; ++i) {
        const int idx = i * 128 + tid;
        const int row = idx >> 4, c4 = idx & 15;
        async_cp16(W + (size_t)(nTile0 + row) * K + kc * 64 + c4 * 4,
                   sW + row * LDS_STRIDE + c4 * 4);
    }
    // A chunk: 16 rows x 64 floats = 256 float4, 2 per thread
    #pragma unroll
    for (int i = 0; i < 2; ++i) {
        const int idx = i * 128 + tid;
        const int row = idx >> 4, c4 = idx & 15;
        async_cp16(A + (size_t)(mTile + row) * K + kc * 64 + c4 * 4,
                   sA + row * LDS_STRIDE + c4 * 4);
    }
}

__global__ void __launch_bounds__(128) wmma_gemm_f32(
    const float* __restrict__ A, const float* __restrict__ W,
    const float* __restrict__ bias, const float* __restrict__ resid,
    float* __restrict__ C, int M, int N, int K, float scale)
{
    const int tid   = threadIdx.x;
    const int lane  = tid & 31;
    const int wave  = tid >> 5;
    const int nTile0 = blockIdx.x * 64;
    const int mTile  = blockIdx.y * 16;
    const int r  = lane & 15;
    const int hi = lane >> 4;

    __shared__ float sA[2][16 * LDS_STRIDE];
    __shared__ float sW[2][64 * LDS_STRIDE];

    const int nCh = K >> 6;       // K / 64
    v8f acc = {};

#ifdef USE_ASYNC_LDS
    stage_chunk(A, W, K, mTile, nTile0, 0, sA[0], sW[0], tid);
    for (int kc = 0; kc < nCh; ++kc) {
        const int cur = kc & 1;
        if (kc + 1 < nCh) {
            stage_chunk(A, W, K, mTile, nTile0, kc + 1, sA[cur ^ 1], sW[cur ^ 1], tid);
            wait_async<10>();   // this chunk's 10 copies done; next 10 in flight
        } else {
            wait_async<0>();
        }
        __syncthreads();
        const float* aP = sA[cur] + r * LDS_STRIDE + hi * 2;
        const float* wP = sW[cur] + (wave * 16 + r) * LDS_STRIDE + hi * 2;
        #pragma unroll
        for (int kk = 0; kk < 64; kk += 4) {
            const v2f a = *(const v2f*)(aP + kk);
            const v2f b = *(const v2f*)(wP + kk);
            acc = __builtin_amdgcn_wmma_f32_16x16x4_f32(
                false, a, false, b, (short)0, acc, false, false);
        }
        __syncthreads();        // buffer consumed before it is re-filled
    }
#else
    for (int kc = 0; kc < nCh; ++kc) {
        stage_chunk(A, W, K, mTile, nTile0, kc, sA[0], sW[0], tid);
        __syncthreads();
        const float* aP = sA[0] + r * LDS_STRIDE + hi * 2;
        const float* wP = sW[0] + (wave * 16 + r) * LDS_STRIDE + hi * 2;
        #pragma unroll
        for (int kk = 0; kk < 64; kk += 4) {
            const v2f a = *(const v2f*)(aP + kk);
            const v2f b = *(const v2f*)(wP + kk);
            acc = __builtin_amdgcn_wmma_f32_16x16x4_f32(
                false, a, false, b, (short)0, acc, false, false);
        }
        __syncthreads();
    }
#endif

    const int n = nTile0 + wave * 16 + r;
    const float bval = bias ? bias[n] : 0.f;
    #pragma unroll
    for (int i = 0; i < 8; ++i) {
        const int m = mTile + i + hi * 8;
        float v = (acc[i] + bval) * scale;
        if (resid) v += resid[(size_t)m * N + n];
        C[(size_t)m * N + n] = v;
    }
}

// ---------------------------------------------------------------------------
// Causal Conv1d(1,1,KER=4) along the feature axis (left-pad 3) + SiLU.
// ---------------------------------------------------------------------------
__global__ void __launch_bounds__(256) conv_silu_k(
    const float* __restrict__ xt, const float* __restrict__ cw,
    const float* __restrict__ cb, float* __restrict__ xc)
{
    const int idx = blockIdx.x * 256 + threadIdx.x;
    const int b = idx / P_SZ;
    const int p = idx - b * P_SZ;
    const float* xr = xt + (size_t)b * P_SZ;
    float acc = cb[0];
    #pragma unroll
    for (int j = 0; j < 4; ++j) {
        const int src = p - 3 + j;
        if (src >= 0) acc += cw[j] * xr[src];
    }
    xc[idx] = acc / (1.f + __expf(-acc));  // silu
}

// ---------------------------------------------------------------------------
// i/f gates: [B,P] x [P,8] -> [B,8]. N=8 < 16 so no WMMA tile; block per b,
// strided-K partials, wave shuffle reduce + LDS atomic combine.
// ---------------------------------------------------------------------------
__global__ void __launch_bounds__(256) if_gemm_k(
    const float* __restrict__ xc,
    const float* __restrict__ Wi, const float* __restrict__ bi,
    const float* __restrict__ Wf, const float* __restrict__ bf,
    float* __restrict__ it, float* __restrict__ ft)
{
    const int b = blockIdx.x;
    const int tid = threadIdx.x;
    const float* xr = xc + (size_t)b * P_SZ;
    float pi[8] = {}, pf[8] = {};
    for (int k = tid; k < P_SZ; k += 256) {
        const float xv = xr[k];
        #pragma unroll
        for (int h = 0; h < 8; ++h) {
            pi[h] += xv * Wi[h * P_SZ + k];
            pf[h] += xv * Wf[h * P_SZ + k];
        }
    }
    #pragma unroll
    for (int h = 0; h < 8; ++h) {
        #pragma unroll
        for (int off = 16; off; off >>= 1) {
            pi[h] += __shfl_xor(pi[h], off, 32);
            pf[h] += __shfl_xor(pf[h], off, 32);
        }
    }
    __shared__ float sI[8], sF[8];
    if (tid < 8)             sI[tid] = 0.f;
    else if (tid < 16)       sF[tid - 8] = 0.f;
    __syncthreads();
    if ((tid & 31) == 0) {
        #pragma unroll
        for (int h = 0; h < 8; ++h) {
            atomicAdd(&sI[h], pi[h]);
            atomicAdd(&sF[h], pf[h]);
        }
    }
    __syncthreads();
    if (tid < 8) {
        it[b * 8 + tid] = sI[tid] + bi[tid];
        ft[b * 8 + tid] = sF[tid] + bf[tid];
    }
}

// ---------------------------------------------------------------------------
// Recurrent update: one block per (b,h). Streams the 128x128 c tile once
// (float4 coalesced), fuses n/m updates, readout, GroupNorm, skip and the
// SiLU(r) gate. Writes c_t/n_t/m_t into d_out regions and y2 into workspace.
// ---------------------------------------------------------------------------
__global__ void __launch_bounds__(256) recurrent_k(
    const float* __restrict__ c_tm1, const float* __restrict__ n_tm1,
    const float* __restrict__ m_tm1, const float* __restrict__ it,
    const float* __restrict__ ft,   const float* __restrict__ q,
    const float* __restrict__ kk,   const float* __restrict__ vv,
    const float* __restrict__ opre, const float* __restrict__ skp,
    const float* __restrict__ rt,   const float* __restrict__ gng,
    const float* __restrict__ gnb,
    float* __restrict__ c_out, float* __restrict__ n_out,
    float* __restrict__ m_out, float* __restrict__ y2)
{
    const int bh = blockIdx.x;
    const int b = bh >> 3, h = bh & 7;
    const int tid = threadIdx.x, wave = tid >> 5, lane = tid & 31;

    __shared__ __align__(16) float sQ[128], sK[128];
    __shared__ float sV[128], sH[128], sRed[128], sScal[2];

    if (tid == 0) {
        const float i0 = it[bh], f0 = ft[bh], mp = m_tm1[bh];
        const float mt = fmaxf(f0 + mp, i0);
        sScal[0] = __expf(i0 - mt);
        sScal[1] = __expf(f0 - mt + mp);
        m_out[bh] = mt;
    }
    const int fb = b * HD_SZ + h * D_SZ;
    if (tid < 128) {
        sQ[tid] = q[fb + tid];
        sK[tid] = kk[fb + tid];   // already scaled by 1/sqrt(D)
        sV[tid] = vv[fb + tid];
    }
    __syncthreads();
    const float ie = sScal[0], fe = sScal[1];

    // n_t and denominator = max(n_t . q_t, 1)
    if (tid < 128) {
        const float nt = fe * n_tm1[(size_t)bh * 128 + tid] + ie * sK[tid];
        n_out[(size_t)bh * 128 + tid] = nt;
        sRed[tid] = nt * sQ[tid];
    }
    __syncthreads();
    for (int s = 64; s > 0; s >>= 1) {
        if (tid < s) sRed[tid] += sRed[tid + s];
        __syncthreads();
    }
    const float denom = fmaxf(sRed[0], 1.0f);
    __syncthreads();

    // c state update + readout numerator, fully coalesced float4 traffic.
    const float4 q4 = ((const float4*)sQ)[lane];
    const float4 k4 = ((const float4*)sK)[lane];
    const size_t base = (size_t)bh * (D_SZ * D_SZ);
    #pragma unroll 4
    for (int pass = 0; pass < 16; ++pass) {
        const int d = pass * 8 + wave;
        const float4 co = ((const float4*)(c_tm1 + base + d * D_SZ))[lane];
        const float ivd = ie * sV[d];
        float4 cn;
        cn.x = fe * co.x + ivd * k4.x;
        cn.y = fe * co.y + ivd * k4.y;
        cn.z = fe * co.z + ivd * k4.z;
        cn.w = fe * co.w + ivd * k4.w;
        ((float4*)(c_out + base + d * D_SZ))[lane] = cn;
        float acc = cn.x * q4.x + cn.y * q4.y + cn.z * q4.z + cn.w * q4.w;
        #pragma unroll
        for (int off = 16; off; off >>= 1) acc += __shfl_xor(acc, off, 32);
        if (lane == 0) sH[d] = acc;
    }
    __syncthreads();

    // h_t = sigmoid(o) * (c_t.q / denom), then GroupNorm over D.
    float htv = 0.f;
    if (tid < 128) {
        const float o  = opre[fb + tid];
        const float sg = 1.f / (1.f + __expf(-o));
        htv = sg * sH[tid] / denom;
        sRed[tid] = htv;
    }
    __syncthreads();
    for (int s = 64; s > 0; s >>= 1) {
        if (tid < s) sRed[tid] += sRed[tid + s];
        __syncthreads();
    }
    const float mu = sRed[0] * (1.f / D_SZ);
    __syncthreads();
    if (tid < 128) { const float d0 = htv - mu; sRed[tid] = d0 * d0; }
    __syncthreads();
    for (int s = 64; s > 0; s >>= 1) {
        if (tid < s) sRed[tid] += sRed[tid + s];
        __syncthreads();
    }
    const float inv = rsqrtf(sRed[0] * (1.f / D_SZ) + 1e-5f);
    if (tid < 128) {
        float yv = (htv - mu) * inv * gng[h * D_SZ + tid] + gnb[h * D_SZ + tid]
                 + skp[fb + tid];
        const float r = rt[fb + tid];
        yv *= r / (1.f + __expf(-r));   // * silu(r_t)
        y2[fb + tid] = yv;
    }
}

// ---------------------------------------------------------------------------
// Host-side launcher.
// ---------------------------------------------------------------------------
extern "C" void kernel_launch(void* const* d_in, const int* in_sizes, int n_in,
                              void* d_out, int out_size, void* d_ws, size_t ws_size,
                              hipStream_t stream)
{
    const float* seq    = (const float*)d_in[0];
    const float* c_tm1  = (const float*)d_in[1];
    const float* n_tm1  = (const float*)d_in[2];
    const float* m_tm1  = (const float*)d_in[3];
    const float* ln_g   = (const float*)d_in[4];
    const float* ln_b   = (const float*)d_in[5];
    const float* Wul    = (const float*)d_in[6];
    const float* bul    = (const float*)d_in[7];
    const float* Wur    = (const float*)d_in[8];
    const float* bur    = (const float*)d_in[9];
    const float* conv_w = (const float*)d_in[10];
    const float* conv_b = (const float*)d_in[11];
    const float* Wskip  = (const float*)d_in[12];
    const float* Wi     = (const float*)d_in[13];
    const float* bi     = (const float*)d_in[14];
    const float* Wf     = (const float*)d_in[15];
    const float* bf     = (const float*)d_in[16];
    const float* Wo     = (const float*)d_in[17];
    const float* bo     = (const float*)d_in[18];
    const float* Wq     = (const float*)d_in[19];
    const float* bq     = (const float*)d_in[20];
    const float* Wk     = (const float*)d_in[21];
    const float* bk     = (const float*)d_in[22];
    const float* Wv     = (const float*)d_in[23];
    const float* bv     = (const float*)d_in[24];
    const float* gn_g   = (const float*)d_in[25];
    const float* gn_b   = (const float*)d_in[26];
    const float* Wd     = (const float*)d_in[27];
    const float* bd     = (const float*)d_in[28];

    // d_out layout: [out 256*1024][c_t 256*8*128*128][n_t 256*8*128][m_t 256*8]
    float* out_m = (float*)d_out;
    float* c_out = out_m + (size_t)B_SZ * DIM_SZ;
    float* n_out = c_out + (size_t)B_SZ * H_SZ * D_SZ * D_SZ;
    float* m_out = n_out + (size_t)B_SZ * H_SZ * D_SZ;

    // workspace layout (floats)
    float* ws    = (float*)d_ws;
    float* x_n   = ws;                                  // 256*1024
    float* x_t   = x_n  + (size_t)B_SZ * DIM_SZ;        // 256*2048
    float* r_t   = x_t  + (size_t)B_SZ * P_SZ;          // 256*1024
    float* x_c   = r_t  + (size_t)B_SZ * HD_SZ;         // 256*2048
    float* q_t   = x_c  + (size_t)B_SZ * P_SZ;          // 256*1024
    float* k_t   = q_t  + (size_t)B_SZ * HD_SZ;         // 256*1024
    float* v_t   = k_t  + (size_t)B_SZ * HD_SZ;         // 256*1024
    float* o_pre = v_t  + (size_t)B_SZ * HD_SZ;         // 256*1024
    float* skp   = o_pre + (size_t)B_SZ * HD_SZ;        // 256*1024
    float* y2    = skp  + (size_t)B_SZ * HD_SZ;         // 256*1024
    float* i_t   = y2   + (size_t)B_SZ * HD_SZ;         // 256*8
    float* f_t   = i_t  + (size_t)B_SZ * H_SZ;          // 256*8

    const float kscale = 0.08838834764831845f;          // 1/sqrt(128)

    // 1. LayerNorm
    layernorm_k<<<B_SZ, 256, 0, stream>>>(seq, ln_g, ln_b, x_n);

    // 2. x_t = x_n @ Wul^T + bul ; r_t = x_n @ Wur^T + bur
    wmma_gemm_f32<<<dim3(P_SZ / 64, B_SZ / 16), 128, 0, stream>>>(
        x_n, Wul, bul, nullptr, x_t, B_SZ, P_SZ, DIM_SZ, 1.f);
    wmma_gemm_f32<<<dim3(HD_SZ / 64, B_SZ / 16), 128, 0, stream>>>(
        x_n, Wur, bur, nullptr, r_t, B_SZ, HD_SZ, DIM_SZ, 1.f);

    // 3. x_c = silu(causal_conv(x_t))
    conv_silu_k<<<(B_SZ * P_SZ) / 256, 256, 0, stream>>>(x_t, conv_w, conv_b, x_c);

    // 4. q/k/skip from x_c ; v/o from x_t  (all K = P = 2048)
    wmma_gemm_f32<<<dim3(HD_SZ / 64, B_SZ / 16), 128, 0, stream>>>(
        x_c, Wq, bq, nullptr, q_t, B_SZ, HD_SZ, P_SZ, 1.f);
    wmma_gemm_f32<<<dim3(HD_SZ / 64, B_SZ / 16), 128, 0, stream>>>(
        x_c, Wk, bk, nullptr, k_t, B_SZ, HD_SZ, P_SZ, kscale);
    wmma_gemm_f32<<<dim3(HD_SZ / 64, B_SZ / 16), 128, 0, stream>>>(
        x_c, Wskip, nullptr, nullptr, skp, B_SZ, HD_SZ, P_SZ, 1.f);
    wmma_gemm_f32<<<dim3(HD_SZ / 64, B_SZ / 16), 128, 0, stream>>>(
        x_t, Wv, bv, nullptr, v_t, B_SZ, HD_SZ, P_SZ, 1.f);
    wmma_gemm_f32<<<dim3(HD_SZ / 64, B_SZ / 16), 128, 0, stream>>>(
        x_t, Wo, bo, nullptr, o_pre, B_SZ, HD_SZ, P_SZ, 1.f);

    // 5. i/f gates (N=8)
    if_gemm_k<<<B_SZ, 256, 0, stream>>>(x_c, Wi, bi, Wf, bf, i_t, f_t);

    // 6. recurrent state update + readout + groupnorm + gating
    recurrent_k<<<B_SZ * H_SZ, 256, 0, stream>>>(
        c_tm1, n_tm1, m_tm1, i_t, f_t, q_t, k_t, v_t, o_pre, skp, r_t,
        gn_g, gn_b, c_out, n_out, m_out, y2);

    // 7. out = y2 @ Wd^T + bd + seq
    wmma_gemm_f32<<<dim3(DIM_SZ / 64, B_SZ / 16), 128, 0, stream>>>(
        y2, Wd, bd, seq, out_m, B_SZ, DIM_SZ, DIM_SZ, 1.f);
}